// GNN_MoE_node_69415261438524
// MI455X (gfx1250) — compile-verified
//
#include <hip/hip_runtime.h>

// ---------------------------------------------------------------------------
// GIN-MoE forward for MI455X (gfx1250, wave32).
// GEMMs use V_WMMA_F32_16X16X4_F32 (fp32 matrix pipe) to match the fp32
// reference numerics. Gather/scatter stays resident in the 192MB L2
// (h/aggr = 25.6MB each). Scatter + cross-block stat combine use hardware
// f32 atomics (global_atomic_add_f32 via unsafeAtomicAdd).
// ---------------------------------------------------------------------------

typedef __attribute__((ext_vector_type(2))) float v2f;
typedef __attribute__((ext_vector_type(8))) float v8f;

#define NNODES 50000
#define NEDGES 800000
#define EMB 128
#define HID 256

// -------------------------------------------------------------- zero scratch
__global__ __launch_bounds__(256) void k_zero(float* __restrict__ p, int n) {
  int i = blockIdx.x * 256 + threadIdx.x;
  if (i < n) p[i] = 0.0f;
}

// ------------------------------------------------------------- atom encoder
__global__ __launch_bounds__(256) void k_atom(
    const int* __restrict__ x,
    const float* __restrict__ e0, const float* __restrict__ e1,
    const float* __restrict__ e2, const float* __restrict__ e3,
    const float* __restrict__ e4, const float* __restrict__ e5,
    const float* __restrict__ e6, const float* __restrict__ e7,
    const float* __restrict__ e8, float* __restrict__ h) {
  int i = blockIdx.x * 256 + threadIdx.x;   // over NNODES*32 float4 slots
  if (i >= NNODES * 32) return;
  int r = i >> 5;
  int c = (i & 31) * 4;
  const int* xr = x + r * 9;
  float4 a = *(const float4*)(e0 + xr[0] * EMB + c);
#define ACC_TAB(T, J)                                        \
  {                                                          \
    float4 v = *(const float4*)(T + xr[J] * EMB + c);        \
    a.x += v.x; a.y += v.y; a.z += v.z; a.w += v.w;          \
  }
  ACC_TAB(e1, 1) ACC_TAB(e2, 2) ACC_TAB(e3, 3) ACC_TAB(e4, 4)
  ACC_TAB(e5, 5) ACC_TAB(e6, 6) ACC_TAB(e7, 7) ACC_TAB(e8, 8)
#undef ACC_TAB
  *(float4*)(h + r * EMB + c) = a;
}

// --------------------------------------- edge message + scatter-add (1 edge/wave)
__global__ __launch_bounds__(256) void k_edges(
    const int* __restrict__ ei, const int* __restrict__ ea,
    const float* __restrict__ be0, const float* __restrict__ be1,
    const float* __restrict__ be2, const float* __restrict__ h,
    float* __restrict__ aggr) {
  int lane = threadIdx.x & 31;
  int e = blockIdx.x * 8 + (threadIdx.x >> 5);
  if (e >= NEDGES) return;
  int s = ei[e];
  int d = ei[NEDGES + e];
  int a0 = ea[e * 3 + 0], a1 = ea[e * 3 + 1], a2 = ea[e * 3 + 2];
  int c = lane * 4;
  float4 hv = *(const float4*)(h + s * EMB + c);
  float4 v0 = *(const float4*)(be0 + a0 * EMB + c);
  float4 v1 = *(const float4*)(be1 + a1 * EMB + c);
  float4 v2 = *(const float4*)(be2 + a2 * EMB + c);
  float m0 = fmaxf(hv.x + v0.x + v1.x + v2.x, 0.0f);
  float m1 = fmaxf(hv.y + v0.y + v1.y + v2.y, 0.0f);
  float m2 = fmaxf(hv.z + v0.z + v1.z + v2.z, 0.0f);
  float m3 = fmaxf(hv.w + v0.w + v1.w + v2.w, 0.0f);
  float* ap = aggr + d * EMB + c;
  unsafeAtomicAdd(ap + 0, m0);
  unsafeAtomicAdd(ap + 1, m1);
  unsafeAtomicAdd(ap + 2, m2);
  unsafeAtomicAdd(ap + 3, m3);
}

// --------------------------------------------------------------------------
// GEMM1: z1[N x 256] = ((1+eps)*h + aggr) @ W1 + b1     (K = 128)
// Block: 256 thr = 8 waves, 32 rows x 256 cols. Wave = 16 rows x 64 cols
// (4 accumulators). A staged in LDS (pad to 130 floats -> conflict free).
// WMMA f32 16x16x4 layouts per ISA 7.12.2:
//   A: lanes 0-15 K={k0,k0+1}, lanes 16-31 K={k0+2,k0+3}; B symmetric;
//   C/D: VGPR v holds M=v (lanes 0-15) / M=v+8 (lanes 16-31).
// --------------------------------------------------------------------------
__global__ __launch_bounds__(256) void k_gemm1(
    const float* __restrict__ h, const float* __restrict__ aggr,
    const float* __restrict__ W, const float* __restrict__ bias,
    const float* __restrict__ epsp, int le, float* __restrict__ z1) {
  __shared__ float As[32][EMB + 2];
  int tid = threadIdx.x;
  int lane = tid & 31;
  int wave = tid >> 5;
  int rowBase = blockIdx.x * 32;
  float ep = 1.0f + epsp[le];
  for (int i = tid; i < 32 * EMB; i += 256) {
    int r = i >> 7, c = i & (EMB - 1);
    int gr = rowBase + r;
    float v = 0.0f;
    if (gr < NNODES) v = ep * h[gr * EMB + c] + aggr[gr * EMB + c];
    As[r][c] = v;
  }
  __syncthreads();
  int wr = wave >> 2;          // row-tile 0/1
  int n0 = (wave & 3) * 64;    // 64 cols per wave
  int nl = lane & 15;
  int kh = (lane < 16) ? 0 : 2;
  int aRow = wr * 16 + nl;
  v8f acc0 = {}, acc1 = {}, acc2 = {}, acc3 = {};
  for (int k0 = 0; k0 < EMB; k0 += 4) {
    v2f a;
    a.x = As[aRow][k0 + kh];
    a.y = As[aRow][k0 + kh + 1];
    const float* Wp = W + (k0 + kh) * HID;
    v2f b0, b1, b2, b3;
    b0.x = Wp[n0 + nl];       b0.y = Wp[HID + n0 + nl];
    b1.x = Wp[n0 + 16 + nl];  b1.y = Wp[HID + n0 + 16 + nl];
    b2.x = Wp[n0 + 32 + nl];  b2.y = Wp[HID + n0 + 32 + nl];
    b3.x = Wp[n0 + 48 + nl];  b3.y = Wp[HID + n0 + 48 + nl];
    acc0 = __builtin_amdgcn_wmma_f32_16x16x4_f32(false, a, false, b0, (short)0, acc0, false, false);
    acc1 = __builtin_amdgcn_wmma_f32_16x16x4_f32(false, a, false, b1, (short)0, acc1, false, false);
    acc2 = __builtin_amdgcn_wmma_f32_16x16x4_f32(false, a, false, b2, (short)0, acc2, false, false);
    acc3 = __builtin_amdgcn_wmma_f32_16x16x4_f32(false, a, false, b3, (short)0, acc3, false, false);
  }
  v8f accs[4] = {acc0, acc1, acc2, acc3};
  int rOff = (lane < 16) ? 0 : 8;
#pragma unroll
  for (int t = 0; t < 4; ++t) {
    int n = n0 + t * 16 + nl;
    float bv = bias[n];
#pragma unroll
    for (int v = 0; v < 8; ++v) {
      int gr = rowBase + wr * 16 + rOff + v;
      if (gr < NNODES) z1[gr * HID + n] = accs[t][v] + bv;
    }
  }
}

// --------------------------------------------------------------------------
// GEMM2: z2[N x 128] = relu(BN(z1)) @ W2 + b2          (K = 256)
// Same scheme; wave = 16 rows x 32 cols (2 accumulators).
// --------------------------------------------------------------------------
__global__ __launch_bounds__(256) void k_gemm2(
    const float* __restrict__ z1, const float* __restrict__ mu,
    const float* __restrict__ rs, const float* __restrict__ g,
    const float* __restrict__ bb, const float* __restrict__ W,
    const float* __restrict__ bias, float* __restrict__ z2) {
  __shared__ float As[32][HID + 2];
  int tid = threadIdx.x;
  int lane = tid & 31;
  int wave = tid >> 5;
  int rowBase = blockIdx.x * 32;
  for (int i = tid; i < 32 * HID; i += 256) {
    int r = i >> 8, c = i & (HID - 1);
    int gr = rowBase + r;
    float v = 0.0f;
    if (gr < NNODES) {
      float z = z1[gr * HID + c];
      v = fmaxf((z - mu[c]) * rs[c] * g[c] + bb[c], 0.0f);
    }
    As[r][c] = v;
  }
  __syncthreads();
  int wr = wave >> 2;
  int n0 = (wave & 3) * 32;
  int nl = lane & 15;
  int kh = (lane < 16) ? 0 : 2;
  int aRow = wr * 16 + nl;
  v8f acc0 = {}, acc1 = {};
  for (int k0 = 0; k0 < HID; k0 += 4) {
    v2f a;
    a.x = As[aRow][k0 + kh];
    a.y = As[aRow][k0 + kh + 1];
    const float* Wp = W + (k0 + kh) * EMB;
    v2f b0, b1;
    b0.x = Wp[n0 + nl];       b0.y = Wp[EMB + n0 + nl];
    b1.x = Wp[n0 + 16 + nl];  b1.y = Wp[EMB + n0 + 16 + nl];
    acc0 = __builtin_amdgcn_wmma_f32_16x16x4_f32(false, a, false, b0, (short)0, acc0, false, false);
    acc1 = __builtin_amdgcn_wmma_f32_16x16x4_f32(false, a, false, b1, (short)0, acc1, false, false);
  }
  v8f accs[2] = {acc0, acc1};
  int rOff = (lane < 16) ? 0 : 8;
#pragma unroll
  for (int t = 0; t < 2; ++t) {
    int n = n0 + t * 16 + nl;
    float bv = bias[n];
#pragma unroll
    for (int v = 0; v < 8; ++v) {
      int gr = rowBase + wr * 16 + rOff + v;
      if (gr < NNODES) z2[gr * EMB + n] = accs[t][v] + bv;
    }
  }
}

// ------------------------------------------- column sum / sumsq for batchnorm
template <int C>
__global__ void k_stats(const float* __restrict__ z, float* __restrict__ sums) {
  int c = threadIdx.x;                 // blockDim == C -> coalesced rows
  int r0 = blockIdx.x * 128;
  float s = 0.0f, s2 = 0.0f;
  for (int i = 0; i < 128; ++i) {
    int r = r0 + i;
    if (r < NNODES) {
      float v = z[r * C + c];
      s += v;
      s2 += v * v;
    }
  }
  unsafeAtomicAdd(&sums[c], s);
  unsafeAtomicAdd(&sums[C + c], s2);
}

__global__ void k_statsfin(const float* __restrict__ sums, float* __restrict__ mu,
                           float* __restrict__ rs, int C) {
  int c = threadIdx.x;
  if (c >= C) return;
  float m = sums[c] * (1.0f / (float)NNODES);
  float var = sums[C + c] * (1.0f / (float)NNODES) - m * m;
  mu[c] = m;
  rs[c] = rsqrtf(fmaxf(var, 0.0f) + 1e-5f);
}

// --------------------------------- BN2 + accumulate expert mean into h_acc
__global__ __launch_bounds__(256) void k_accum(
    const float* __restrict__ z2, const float* __restrict__ mu,
    const float* __restrict__ rs, const float* __restrict__ g,
    const float* __restrict__ bb, float* __restrict__ hacc) {
  int i = blockIdx.x * 256 + threadIdx.x;
  if (i >= NNODES * EMB) return;
  int c = i & (EMB - 1);
  float v = (z2[i] - mu[c]) * rs[c] * g[c] + bb[c];
  hacc[i] += v * (1.0f / 3.0f);
}

// --------------------------------------------- layer finalize (optional relu)
__global__ __launch_bounds__(256) void k_final(const float* __restrict__ hacc,
                                               float* __restrict__ hout,
                                               int doRelu) {
  int i = blockIdx.x * 256 + threadIdx.x;
  if (i >= NNODES * EMB) return;
  float v = hacc[i];
  hout[i] = doRelu ? fmaxf(v, 0.0f) : v;
}

// ---------------------------------------------------------------------------
extern "C" void kernel_launch(void* const* d_in, const int* in_sizes, int n_in,
                              void* d_out, int out_size, void* d_ws, size_t ws_size,
                              hipStream_t stream) {
  // setup_inputs() dict order, params lists expanded in insertion order.
  const int* x = (const int*)d_in[0];
  const int* ei = (const int*)d_in[1];
  const int* ea = (const int*)d_in[2];
  const float* ae[9];
  for (int i = 0; i < 9; ++i) ae[i] = (const float*)d_in[3 + i];
  const float* be0 = (const float*)d_in[12];   // [5,3,5,128]
  const float* be1 = (const float*)d_in[13];   // [5,3,6,128]
  const float* be2 = (const float*)d_in[14];   // [5,3,2,128]
  const float* eps = (const float*)d_in[15];   // [5,3]
  const float* W1 = (const float*)d_in[16];    // [5,3,128,256]
  const float* b1 = (const float*)d_in[17];    // [5,3,256]
  const float* g1 = (const float*)d_in[18];
  const float* bb1 = (const float*)d_in[19];
  const float* W2 = (const float*)d_in[20];    // [5,3,256,128]
  const float* b2 = (const float*)d_in[21];    // [5,3,128]
  const float* g2 = (const float*)d_in[22];
  const float* bb2 = (const float*)d_in[23];

  // workspace carve-up (z2 aliases aggr: aggr dead after GEMM1)
  float* h = (float*)d_ws;
  float* hacc = h + NNODES * EMB;
  float* aggr = hacc + NNODES * EMB;
  float* z1 = aggr + NNODES * EMB;
  float* z2 = aggr;
  float* sums1 = z1 + NNODES * HID;
  float* mu1 = sums1 + 2 * HID;
  float* rs1 = mu1 + HID;
  float* sums2 = rs1 + HID;
  float* mu2 = sums2 + 2 * EMB;
  float* rs2 = mu2 + EMB;

  const int nElem = NNODES * EMB;                 // 6.4M
  const int zeroBlk = (nElem + 255) / 256;
  const int gemmBlk = (NNODES + 31) / 32;         // 1563
  const int edgeBlk = (NEDGES + 7) / 8;           // 100000
  const int statBlk = (NNODES + 127) / 128;       // 391
  const int atomBlk = (NNODES * 32 + 255) / 256;

  k_atom<<<atomBlk, 256, 0, stream>>>(x, ae[0], ae[1], ae[2], ae[3], ae[4],
                                      ae[5], ae[6], ae[7], ae[8], h);

  for (int layer = 0; layer < 5; ++layer) {
    float* hout = (layer == 4) ? (float*)d_out : h;
    k_zero<<<zeroBlk, 256, 0, stream>>>(hacc, nElem);
    for (int e = 0; e < 3; ++e) {
      int le = layer * 3 + e;
      k_zero<<<zeroBlk, 256, 0, stream>>>(aggr, nElem);
      k_edges<<<edgeBlk, 256, 0, stream>>>(ei, ea, be0 + (size_t)le * 5 * EMB,
                                           be1 + (size_t)le * 6 * EMB,
                                           be2 + (size_t)le * 2 * EMB, h, aggr);
      k_gemm1<<<gemmBlk, 256, 0, stream>>>(h, aggr, W1 + (size_t)le * EMB * HID,
                                           b1 + (size_t)le * HID, eps, le, z1);
      k_zero<<<2, 256, 0, stream>>>(sums1, 2 * HID);
      k_stats<HID><<<statBlk, HID, 0, stream>>>(z1, sums1);
      k_statsfin<<<1, HID, 0, stream>>>(sums1, mu1, rs1, HID);
      k_gemm2<<<gemmBlk, 256, 0, stream>>>(z1, mu1, rs1, g1 + (size_t)le * HID,
                                           bb1 + (size_t)le * HID,
                                           W2 + (size_t)le * HID * EMB,
                                           b2 + (size_t)le * EMB, z2);
      k_zero<<<1, 256, 0, stream>>>(sums2, 2 * EMB);
      k_stats<EMB><<<statBlk, EMB, 0, stream>>>(z2, sums2);
      k_statsfin<<<1, EMB, 0, stream>>>(sums2, mu2, rs2, EMB);
      k_accum<<<zeroBlk, 256, 0, stream>>>(z2, mu2, rs2, g2 + (size_t)le * EMB,
                                           bb2 + (size_t)le * EMB, hacc);
    }
    k_final<<<zeroBlk, 256, 0, stream>>>(hacc, hout, layer < 4 ? 1 : 0);
  }
}